// RobustGCNLayer_74354473828386
// MI455X (gfx1250) — compile-verified
//
#include <hip/hip_runtime.h>
#include <hip/hip_bf16.h>

typedef float v2f __attribute__((ext_vector_type(2)));
typedef float v8f __attribute__((ext_vector_type(8)));

#define D_DIM 512
#define H_DIM 256
#define LDA   516   // padded LDS row stride (words): 516 % 64 = 4 -> conflict-free frag reads

// ---------------------------------------------------------------------------
// Kernel 0: out[n*H + h] = b[h]   (bias pre-seeded; also clears poison)
// ---------------------------------------------------------------------------
__global__ __launch_bounds__(256) void gcn_init_bias(const float* __restrict__ b,
                                                     float* __restrict__ out,
                                                     int total) {
    int i = blockIdx.x * 256 + threadIdx.x;
    if (i < total) out[i] = b[i & (H_DIM - 1)];
}

// ---------------------------------------------------------------------------
// Kernel 1: xw = x @ W via V_WMMA_F32_16X16X4_F32 (fp32-exact matrix path)
// Block = 256 threads = 8 waves; block computes rows [16*bid, 16*bid+16) x all
// 256 columns. A strip staged in LDS once per block; each wave handles 32
// columns with two accumulators to pipeline the WMMA unit.
// ---------------------------------------------------------------------------
__global__ __launch_bounds__(256) void gcn_gemm_wmma(const float* __restrict__ x,
                                                     const float* __restrict__ W,
                                                     float* __restrict__ xw,
                                                     int N) {
    __shared__ float As[16 * LDA];   // ~33 KB of the 320 KB WGP LDS

    const int row0 = blockIdx.x * 16;

    // Cooperative load of the 16x512 fp32 A strip (8192 floats = 2048 float4).
    // 128 float4 per row; padded row stride keeps 16-B alignment (2064 % 16 == 0).
    for (int i = threadIdx.x; i < 16 * (D_DIM / 4); i += 256) {
        const int r  = i >> 7;          // /128
        const int c4 = i & 127;
        const int gr = row0 + r;
        float4 v;
        if (gr < N) {
            v = ((const float4*)(x + (size_t)gr * D_DIM))[c4];
        } else {
            v = make_float4(0.f, 0.f, 0.f, 0.f);
        }
        *(float4*)(&As[r * LDA + c4 * 4]) = v;
    }
    __syncthreads();

    const int wave  = threadIdx.x >> 5;      // 0..7
    const int lane  = threadIdx.x & 31;
    const int lrow  = lane & 15;             // M (A) / N (B,D) position
    const int khalf = lane >> 4;             // lane-group: K sub-pair select
    const int col0  = wave * 32;             // this wave: columns [col0, col0+32)

    v8f acc0 = {};
    v8f acc1 = {};

    const float* __restrict__ Wp = W + (size_t)(2 * khalf) * H_DIM + lrow;

    #pragma unroll 8
    for (int k = 0; k < D_DIM; k += 4) {
        // A 16x4 fragment (ISA 7.12.2): lane group g holds K = k+2g, k+2g+1 of row lrow
        v2f a = *(const v2f*)(&As[lrow * LDA + k + 2 * khalf]);

        // B 4x16 fragment: VGPR v, lane group g -> K = k + v + 2g, N = lrow
        const float* wk = Wp + (size_t)k * H_DIM;
        v2f b0, b1;
        b0.x = wk[col0];                 // K = k+2g,   N = col0+lrow
        b0.y = wk[H_DIM + col0];         // K = k+2g+1
        b1.x = wk[col0 + 16];            // second 16-wide column tile
        b1.y = wk[H_DIM + col0 + 16];

        acc0 = __builtin_amdgcn_wmma_f32_16x16x4_f32(false, a, false, b0,
                                                     (short)0, acc0, false, false);
        acc1 = __builtin_amdgcn_wmma_f32_16x16x4_f32(false, a, false, b1,
                                                     (short)0, acc1, false, false);
    }

    // D 16x16 layout: VGPR r, lane group g -> M = r + 8g, N = lrow
    float* outp = xw + (size_t)row0 * H_DIM + col0 + lrow;
    #pragma unroll
    for (int r = 0; r < 8; ++r) {
        const int m = r + 8 * khalf;
        if (row0 + m < N) {
            outp[(size_t)m * H_DIM]      = acc0[r];
            outp[(size_t)m * H_DIM + 16] = acc1[r];
        }
    }
}

// ---------------------------------------------------------------------------
// Kernel 2: scatter-add  out[row[e]] += vals[e] * xw[col[e]]
// One wave per edge: 32 lanes x float4 = 128 floats/iter, 2 iters covers H=256.
// Gather is a coalesced 1 KB row read (hot in 192 MB L2); scatter uses
// non-returning global_atomic_add_f32.
// ---------------------------------------------------------------------------
__global__ __launch_bounds__(256) void gcn_spmm_scatter(const float* __restrict__ xw,
                                                        const int* __restrict__ row,
                                                        const int* __restrict__ col,
                                                        const float* __restrict__ vals,
                                                        float* __restrict__ out,
                                                        int E) {
    const int wave = threadIdx.x >> 5;
    const int lane = threadIdx.x & 31;
    const int e    = blockIdx.x * 8 + wave;
    if (e >= E) return;

    const int   r = row[e];
    const int   c = col[e];
    const float v = vals[e];

    const float4* __restrict__ src = (const float4*)(xw + (size_t)c * H_DIM);
    float* __restrict__ dst = out + (size_t)r * H_DIM;

    #pragma unroll
    for (int i = 0; i < 2; ++i) {
        const int q = lane + 32 * i;         // float4 index within the row
        float4 m = src[q];
        const int base = q * 4;
        atomicAdd(dst + base + 0, v * m.x);
        atomicAdd(dst + base + 1, v * m.y);
        atomicAdd(dst + base + 2, v * m.z);
        atomicAdd(dst + base + 3, v * m.w);
    }
}

// ---------------------------------------------------------------------------
extern "C" void kernel_launch(void* const* d_in, const int* in_sizes, int n_in,
                              void* d_out, int out_size, void* d_ws, size_t ws_size,
                              hipStream_t stream) {
    const float* x    = (const float*)d_in[0];
    const float* W    = (const float*)d_in[1];
    const float* b    = (const float*)d_in[2];
    const int*   row  = (const int*)d_in[3];
    const int*   col  = (const int*)d_in[4];
    const float* vals = (const float*)d_in[5];
    float*       out  = (float*)d_out;

    const int N = in_sizes[0] / D_DIM;   // 100000
    const int E = in_sizes[3];           // 3200000

    float* xw = (float*)d_ws;            // N*H fp32 = 102.4 MB scratch

    // 1) out = broadcast(bias)
    {
        const int total = N * H_DIM;
        gcn_init_bias<<<(total + 255) / 256, 256, 0, stream>>>(b, out, total);
    }
    // 2) xw = x @ W  (WMMA f32)
    {
        const int nblk = (N + 15) / 16;
        gcn_gemm_wmma<<<nblk, 256, 0, stream>>>(x, W, xw, N);
    }
    // 3) out += scatter-add of vals * xw[col]
    {
        const int nblk = (E + 7) / 8;
        gcn_spmm_scatter<<<nblk, 256, 0, stream>>>(xw, row, col, vals, out, E);
    }
}